// GlobalPointer_4501125726706
// MI455X (gfx1250) — compile-verified
//
#include <hip/hip_runtime.h>
#include <hip/hip_bf16.h>

#define HIDDEN 1024
#define HEADS 12
#define HEAD_SIZE 64
#define BATCH 8
#define SEQ 512
#define ROWS (BATCH * SEQ)              // 4096
#define OUT_DIM (HEADS * HEAD_SIZE * 2) // 1536
#define NEG_INFF (-1e30f)
#define TRIL_PEN (1e12f)

typedef __attribute__((ext_vector_type(2))) float v2f;
typedef __attribute__((ext_vector_type(8))) float v8f;

// Exact-precision f32 WMMA: D(16x16) = A(16x4) * B(4x16) + C
__device__ __forceinline__ v8f wmma4(v2f a, v2f b, v8f c) {
  return __builtin_amdgcn_wmma_f32_16x16x4_f32(
      /*neg_a=*/false, a, /*neg_b=*/false, b,
      /*c_mod=*/(short)0, c, /*reuse_a=*/false, /*reuse_b=*/false);
}

// ---------------------------------------------------------------------------
// Kernel 1: Y = X @ W^T + bias, split q/k, apply RoPE, write to workspace
// (b, h, seq, d) row-major so kernel 2's fragment loads are contiguous.
// One wave -> 32 (rows) x 64 (one q- or k-block of one head):
// 8 WMMAs per 6 fragment loads per K-step.
// ---------------------------------------------------------------------------
__global__ void __launch_bounds__(128)
qkv_rope_kernel(const float* __restrict__ X, const float* __restrict__ W,
                const float* __restrict__ bias,
                const float* __restrict__ cosp, const float* __restrict__ sinp,
                float* __restrict__ q_ws, float* __restrict__ k_ws) {
  __shared__ float sm_all[4 * 32 * 65];      // 4 waves * 32x64 tile, pad to 65
  const int wv   = threadIdx.x >> 5;
  const int lane = threadIdx.x & 31;
  const int col  = lane & 15;
  const int half = lane >> 4;

  const int gid     = blockIdx.x * 4 + wv;   // 0..3071 tiles
  const int rowTile = gid / 24;              // 128 row tiles of 32
  const int cb      = gid % 24;              // 24 column blocks of 64
  const int m0      = rowTile * 32;
  const int o0      = cb * 64;
  const int h       = cb >> 1;
  const int isK     = cb & 1;

  const float* A0 = X + (size_t)(m0 + col) * HIDDEN + 2 * half;
  const float* A1 = X + (size_t)(m0 + 16 + col) * HIDDEN + 2 * half;
  const float* B0 = W + (size_t)(o0 + col) * HIDDEN + 2 * half;

  v8f acc00 = {}, acc01 = {}, acc10 = {}, acc11 = {};
  v8f acc20 = {}, acc21 = {}, acc30 = {}, acc31 = {};

#pragma unroll 4
  for (int k = 0; k < HIDDEN; k += 4) {
    v2f a0 = *(const v2f*)(A0 + k);
    v2f a1 = *(const v2f*)(A1 + k);
    v2f b0 = *(const v2f*)(B0 + k);
    v2f b1 = *(const v2f*)(B0 + 16 * HIDDEN + k);
    v2f b2 = *(const v2f*)(B0 + 32 * HIDDEN + k);
    v2f b3 = *(const v2f*)(B0 + 48 * HIDDEN + k);
    __builtin_prefetch(A0 + k + 256, 0, 3);   // global_prefetch_b8, near scope
    __builtin_prefetch(B0 + k + 256, 0, 3);
    acc00 = wmma4(a0, b0, acc00);  acc01 = wmma4(a1, b0, acc01);
    acc10 = wmma4(a0, b1, acc10);  acc11 = wmma4(a1, b1, acc11);
    acc20 = wmma4(a0, b2, acc20);  acc21 = wmma4(a1, b2, acc21);
    acc30 = wmma4(a0, b3, acc30);  acc31 = wmma4(a1, b3, acc31);
  }

  // Stage tile (plus bias) in LDS so RoPE can gather partner elements.
  float* sm = sm_all + wv * 32 * 65;
  float bv[4];
#pragma unroll
  for (int nb = 0; nb < 4; ++nb) bv[nb] = bias[o0 + nb * 16 + col];

  v8f accs[4][2] = {{acc00, acc01}, {acc10, acc11}, {acc20, acc21}, {acc30, acc31}};
#pragma unroll
  for (int nb = 0; nb < 4; ++nb)
#pragma unroll
    for (int t = 0; t < 2; ++t)
#pragma unroll
      for (int j = 0; j < 8; ++j) {
        const int r = 16 * t + j + 8 * half;  // C/D layout: VGPR j -> rows j, j+8
        sm[r * 65 + nb * 16 + col] = accs[nb][t][j] + bv[nb];
      }

  float* dst_base = isK ? k_ws : q_ws;
#pragma unroll 4
  for (int r = 0; r < 32; ++r) {
    const int mg   = m0 + r;
    const int b    = mg >> 9;                 // /512
    const int nseq = mg & 511;
    const float* cr  = cosp + nseq * HEAD_SIZE;
    const float* sr  = sinp + nseq * HEAD_SIZE;
    const float* row = sm + r * 65;
    float* dst = dst_base + ((size_t)(b * HEADS + h) * SEQ + nseq) * HEAD_SIZE;
#pragma unroll
    for (int dd = 0; dd < 2; ++dd) {
      const int d = lane + 32 * dd;
      const float x  = row[d];
      // x2[d] = -x[2d+1] (d<32) ; x[2(d-32)] (d>=32)
      const float x2 = (d < 32) ? -row[2 * d + 1] : row[2 * (d - 32)];
      dst[d] = x * cr[d] + x2 * sr[d];
    }
  }
}

// ---------------------------------------------------------------------------
// Kernel 2: logits[b,h,m,n] = (mask? q.k : -1e30) - 1e12*(m>n), then *0.125
// One wave -> 32(M) x 64(N) tile of one (b,h) 512x512 plane.
// K=64 fully unrolled: 128 WMMAs vs 96 fragment loads per wave.
// ---------------------------------------------------------------------------
__global__ void __launch_bounds__(256)
attn_logits_kernel(const float* __restrict__ q_ws, const float* __restrict__ k_ws,
                   const int* __restrict__ mask, float* __restrict__ out) {
  const int wv   = threadIdx.x >> 5;
  const int lane = threadIdx.x & 31;
  const int col  = lane & 15;
  const int half = lane >> 4;

  const int gid = blockIdx.x * 8 + wv;       // 0..12287
  const int bh  = gid >> 7;                  // 96 (b,h) planes
  const int rem = gid & 127;
  const int m0  = (rem >> 3) * 32;           // 16 m-tiles of 32
  const int n0  = (rem & 7) * 64;            // 8 n-strips of 64
  const int b   = bh / HEADS;

  const float* Q = q_ws + (size_t)bh * SEQ * HEAD_SIZE;
  const float* K = k_ws + (size_t)bh * SEQ * HEAD_SIZE;
  const float* QA0 = Q + (size_t)(m0 + col) * HEAD_SIZE + 2 * half;
  const float* QA1 = Q + (size_t)(m0 + 16 + col) * HEAD_SIZE + 2 * half;
  const float* KB0 = K + (size_t)(n0 + col) * HEAD_SIZE + 2 * half;

  v8f acc00 = {}, acc01 = {}, acc10 = {}, acc11 = {};
  v8f acc20 = {}, acc21 = {}, acc30 = {}, acc31 = {};

#pragma unroll
  for (int k = 0; k < HEAD_SIZE; k += 4) {
    v2f a0 = *(const v2f*)(QA0 + k);
    v2f a1 = *(const v2f*)(QA1 + k);
    v2f b0 = *(const v2f*)(KB0 + k);
    v2f b1 = *(const v2f*)(KB0 + 16 * HEAD_SIZE + k);
    v2f b2 = *(const v2f*)(KB0 + 32 * HEAD_SIZE + k);
    v2f b3 = *(const v2f*)(KB0 + 48 * HEAD_SIZE + k);
    acc00 = wmma4(a0, b0, acc00);  acc01 = wmma4(a1, b0, acc01);
    acc10 = wmma4(a0, b1, acc10);  acc11 = wmma4(a1, b1, acc11);
    acc20 = wmma4(a0, b2, acc20);  acc21 = wmma4(a1, b2, acc21);
    acc30 = wmma4(a0, b3, acc30);  acc31 = wmma4(a1, b3, acc31);
  }

  const int* mrow = mask + b * SEQ;
  int mm[2][8];
#pragma unroll
  for (int t = 0; t < 2; ++t)
#pragma unroll
    for (int j = 0; j < 8; ++j) mm[t][j] = mrow[m0 + 16 * t + j + 8 * half];

  v8f accs[4][2] = {{acc00, acc01}, {acc10, acc11}, {acc20, acc21}, {acc30, acc31}};
#pragma unroll
  for (int nb = 0; nb < 4; ++nb) {
    const int n  = n0 + nb * 16 + col;
    const int mn = mrow[n];
#pragma unroll
    for (int t = 0; t < 2; ++t) {
      v8f A = accs[nb][t];
#pragma unroll
      for (int j = 0; j < 8; ++j) {
        const int m = m0 + 16 * t + j + 8 * half;
        float v = A[j];
        if (!(mm[t][j] && mn)) v = NEG_INFF;
        if (m > n) v -= TRIL_PEN;
        out[((size_t)bh * SEQ + m) * SEQ + n] = v * 0.125f;  // 1/sqrt(64)
      }
    }
  }
}

// ---------------------------------------------------------------------------
extern "C" void kernel_launch(void* const* d_in, const int* in_sizes, int n_in,
                              void* d_out, int out_size, void* d_ws, size_t ws_size,
                              hipStream_t stream) {
  const float* X    = (const float*)d_in[0];   // (8,512,1024)
  const int*   mask = (const int*)  d_in[1];   // (8,512)
  const float* W    = (const float*)d_in[2];   // (1536,1024)
  const float* bias = (const float*)d_in[3];   // (1536)
  const float* cosp = (const float*)d_in[4];   // (512,64)
  const float* sinp = (const float*)d_in[5];   // (512,64)
  float* out = (float*)d_out;                  // (8,12,512,512)

  float* q_ws = (float*)d_ws;                                       // 12.6 MB
  float* k_ws = q_ws + (size_t)BATCH * HEADS * SEQ * HEAD_SIZE;     // 12.6 MB

  // 3072 32x64 tiles / 4 waves per 128-thread block
  qkv_rope_kernel<<<768, 128, 0, stream>>>(X, W, bias, cosp, sinp, q_ws, k_ws);
  // 12288 32x64 tiles / 8 waves per 256-thread block
  attn_logits_kernel<<<1536, 256, 0, stream>>>(q_ws, k_ws, mask, out);
}